// CausalSelfAttention_42245298324145
// MI455X (gfx1250) — compile-verified
//
#include <hip/hip_runtime.h>

typedef float v2f __attribute__((ext_vector_type(2)));
typedef float v8f __attribute__((ext_vector_type(8)));

__device__ __forceinline__ v8f wmma_f32_16x16x4(v2f a, v2f b, v8f c) {
    // (neg_a, A, neg_b, B, c_mod, C, reuse_a, reuse_b)
    return __builtin_amdgcn_wmma_f32_16x16x4_f32(false, a, false, b, (short)0, c,
                                                 false, false);
}

// Async global -> LDS copies (tracked by ASYNCcnt). Per-lane LDS byte address
// in a VGPR, 64-bit global address in a VGPR pair.
__device__ __forceinline__ void async_ld128(unsigned lds_byte, const float* g) {
    asm volatile("global_load_async_to_lds_b128 %0, %1, off"
                 :: "v"(lds_byte), "v"(g) : "memory");
}
__device__ __forceinline__ void async_ld32(unsigned lds_byte, const float* g) {
    asm volatile("global_load_async_to_lds_b32 %0, %1, off"
                 :: "v"(lds_byte), "v"(g) : "memory");
}
__device__ __forceinline__ void wait_async0() {
    asm volatile("s_wait_asynccnt 0x0" ::: "memory");
}
// Generic pointers to LDS carry the wave-relative LDS offset in bits [31:0].
__device__ __forceinline__ unsigned lds_off(const void* p) {
    return (unsigned)(unsigned long long)(uintptr_t)p;
}

// ---------------------------------------------------------------------------
// GEMM: C[M,N] = A[M,K] @ B[K,N], fp32 via V_WMMA_F32_16X16X4_F32.
// Block = 256 threads = 8 waves computing a 64x128 tile (wave w owns M-subtile
// w&3 and N-subtiles 4*(w>>2)..+3 -> 4 accumulators).
// K staged in KC=32 chunks, double-buffered LDS filled one compute phase ahead:
//   A: global_load_async_to_lds_b128 into row-major [64][36]  (k-pairs contiguous)
//   B: global_load_async_to_lds_b32  TRANSPOSED into [128][34] (k-pairs contiguous)
// so every WMMA operand is a single aligned ds_load_b64, no register shuffling.
// mode==0: plain row-major store to C. mode==1: scatter to q/k/v [B,H,T,hd].
// ---------------------------------------------------------------------------
#define GKC 32
#define LDA_S 36
#define LDBT_S 34
#define LDSA_B (64 * LDA_S * 4)     // bytes per A buffer
#define LDSB_B (128 * LDBT_S * 4)   // bytes per B^T buffer

__global__ __launch_bounds__(256) void attn_gemm_f32_wmma(
    const float* __restrict__ A, const float* __restrict__ B,
    float* __restrict__ C, int M, int N, int K, int mode,
    float* __restrict__ qb, float* __restrict__ kb, float* __restrict__ vb) {
    __shared__ __attribute__((aligned(16))) float lA[2][64 * LDA_S];
    __shared__ __attribute__((aligned(16))) float lBt[2][128 * LDBT_S];

    const int tid  = threadIdx.x;
    const int wave = tid >> 5;
    const int lane = tid & 31;
    const int lo   = lane & 15;
    const int hi   = lane >> 4;
    const int klo  = hi * 2;

    const int bm = blockIdx.x * 64;
    const int bn = blockIdx.y * 128;
    const int mi  = wave & 3;
    const int nj0 = (wave >> 2) << 2;

    // --- async staging addresses -------------------------------------------
    // A tile 64x32 as 512 float4: thread t -> f4 idx {t, t+256} =
    // rows {ar, ar+32}, same column.
    const int ar = tid >> 3, ac = (tid & 7) << 2;
    const float* Ap0 = A + (size_t)(bm + ar) * K + ac;
    const float* Ap1 = Ap0 + (size_t)32 * K;
    const unsigned laA0 = lds_off(&lA[0][0]) + (unsigned)(ar * LDA_S + ac) * 4u;
    const unsigned laA1 = laA0 + (unsigned)(32 * LDA_S) * 4u;
    // B tile 32x128 staged transposed, 4B granularity: global element (k,n)
    // with k = it*2 + (tid>>7), n = tid&127 -> coalesced row reads, LDS
    // scatter to lBt[n][k].
    const int bk0 = tid >> 7, bnn = tid & 127;
    const float* Bp = B + (size_t)bk0 * N + bn + bnn;
    const unsigned laBT = lds_off(&lBt[0][0]) + (unsigned)(bnn * LDBT_S + bk0) * 4u;

#define STAGE(nb, kcn)                                                        \
    do {                                                                      \
        async_ld128(laA0 + (nb)*LDSA_B, Ap0 + (kcn));                         \
        async_ld128(laA1 + (nb)*LDSA_B, Ap1 + (kcn));                         \
        _Pragma("unroll")                                                     \
        for (int it = 0; it < 16; ++it)                                       \
            async_ld32(laBT + (nb)*LDSB_B + (unsigned)(it * 2) * 4u,          \
                       Bp + (size_t)((kcn) + it * 2) * N);                    \
    } while (0)

    v8f acc[4];
#pragma unroll
    for (int j = 0; j < 4; ++j) acc[j] = (v8f){0.f,0.f,0.f,0.f,0.f,0.f,0.f,0.f};

    STAGE(0, 0);                       // prologue: fill buffer 0
    int buf = 0;
    for (int kc = 0; kc < K; kc += GKC) {
        wait_async0();                 // my copies into `buf` landed
        __syncthreads();               // everyone's copies landed
        if (kc + GKC < K) STAGE(buf ^ 1, kc + GKC);   // prefetch next chunk

        const float* sA  = &lA[buf][0];
        const float* sBt = &lBt[buf][0];
#pragma unroll
        for (int kk = 0; kk < GKC; kk += 4) {
            v2f a = *(const v2f*)&sA[(mi * 16 + lo) * LDA_S + kk + klo];
#pragma unroll
            for (int j = 0; j < 4; ++j) {
                v2f b = *(const v2f*)&sBt[((nj0 + j) * 16 + lo) * LDBT_S + kk + klo];
                acc[j] = wmma_f32_16x16x4(a, b, acc[j]);
            }
        }
        __syncthreads();               // all reads of `buf` done before restage
        buf ^= 1;
    }
#undef STAGE

#pragma unroll
    for (int j = 0; j < 4; ++j) {
        int coln = bn + (nj0 + j) * 16 + lo;
#pragma unroll
        for (int r = 0; r < 8; ++r) {
            int row = bm + mi * 16 + r + hi * 8;
            float val = acc[j][r];
            if (mode == 0) {
                C[(size_t)row * N + coln] = val;
            } else {
                int which = coln >> 11;      // 0=q,1=k,2=v
                int c = coln & 2047;
                int h = c >> 5, d = c & 31;
                int b = row >> 10, t = row & 1023;
                float* dst = (which == 0) ? qb : ((which == 1) ? kb : vb);
                dst[((((size_t)b * 64 + h) * 1024 + t) << 5) + d] = val;
            }
        }
    }
}

// ---------------------------------------------------------------------------
// RoPE, in place on q/k laid out [B,H,T,32]. Module quirk: cos/sin index is
// the element index d (0..31), theta_d = pos[t] * 10000^(-d/32).
// ---------------------------------------------------------------------------
__global__ __launch_bounds__(256) void rope_inplace(
    float* __restrict__ qb, float* __restrict__ kb,
    const float* __restrict__ pos) {
    int idx = blockIdx.x * blockDim.x + threadIdx.x;   // B*H*T*16 pairs
    const int total = 2 * 64 * 1024 * 16;
    if (idx >= total) return;
    int i   = idx & 15;
    int t   = (idx >> 4) & 1023;
    size_t base = ((size_t)(idx >> 4)) * 32 + 2 * i;
    float ft = pos[t];
    float f0 = powf(10000.0f, -(float)(2 * i) / 32.0f);
    float f1 = powf(10000.0f, -(float)(2 * i + 1) / 32.0f);
    float c0 = cosf(ft * f0), s0 = sinf(ft * f0);
    float c1 = cosf(ft * f1), s1 = sinf(ft * f1);
    float2 q = *(float2*)&qb[base];
    float2 k = *(float2*)&kb[base];
    float2 qn, kn;
    qn.x = q.x * c0 - q.y * s0;  qn.y = q.y * c1 + q.x * s1;
    kn.x = k.x * c0 - k.y * s0;  kn.y = k.y * c1 + k.x * s1;
    *(float2*)&qb[base] = qn;
    *(float2*)&kb[base] = kn;
}

// ---------------------------------------------------------------------------
// Flash attention, fp32 WMMA. One wave per (b,h,q-tile16). T=1024, hd=32.
// S = QK^T via 8 wmma; online softmax with shfl_xor row reductions (C-layout
// row = vgpr r + hi*8, xor masks 1..8 reduce exactly one row); P converted
// C->A layout through a wave-private LDS tile (DS ops are in-order per wave);
// P@V via 8 wmma into two 16x16 accumulators (hd split into 2 N-tiles).
// ---------------------------------------------------------------------------
#define LP_S 18   // pad: even (8B-aligned v2f reads) and bank-spread

__global__ __launch_bounds__(128) void flash_attn_f32_wmma(
    const float* __restrict__ qb, const float* __restrict__ kb,
    const float* __restrict__ vb, float* __restrict__ y) {
    __shared__ float lP[4 * 16 * LP_S];

    const int tid  = threadIdx.x;
    const int wave = tid >> 5;
    const int lane = tid & 31;
    const int lo   = lane & 15;
    const int hi   = lane >> 4;
    const int klo  = hi * 2;

    const int gw = blockIdx.x * 4 + wave;   // 0..8191
    const int qi = gw & 63;
    const int bh = gw >> 6;                 // b*64+h
    const float* Q  = qb + (size_t)bh * 1024 * 32;
    const float* Km = kb + (size_t)bh * 1024 * 32;
    const float* V  = vb + (size_t)bh * 1024 * 32;
    float* lp = &lP[wave * 16 * LP_S];

    v2f qa[8];
    const int qrow = qi * 16 + lo;
#pragma unroll
    for (int s = 0; s < 8; ++s)
        qa[s] = *(const v2f*)&Q[(size_t)qrow * 32 + s * 4 + klo];

    v8f o0 = {0.f, 0.f, 0.f, 0.f, 0.f, 0.f, 0.f, 0.f};
    v8f o1 = {0.f, 0.f, 0.f, 0.f, 0.f, 0.f, 0.f, 0.f};
    float mrow[8], lrow[8];
#pragma unroll
    for (int r = 0; r < 8; ++r) { mrow[r] = -1e30f; lrow[r] = 0.f; }
    const float scale = 0.17677669529663687f;   // 1/sqrt(32)

    for (int kj = 0; kj <= qi; ++kj) {
        v8f s = {0.f, 0.f, 0.f, 0.f, 0.f, 0.f, 0.f, 0.f};
#pragma unroll
        for (int t = 0; t < 8; ++t) {
            v2f kf = *(const v2f*)&Km[(size_t)(kj * 16 + lo) * 32 + t * 4 + klo];
            s = wmma_f32_16x16x4(qa[t], kf, s);
        }
        const bool diag = (kj == qi);
#pragma unroll
        for (int r = 0; r < 8; ++r) {
            float sv = s[r] * scale;
            int m = r + hi * 8;
            if (diag && lo > m) sv = -1e30f;
            float mx = sv;
            mx = fmaxf(mx, __shfl_xor(mx, 1));
            mx = fmaxf(mx, __shfl_xor(mx, 2));
            mx = fmaxf(mx, __shfl_xor(mx, 4));
            mx = fmaxf(mx, __shfl_xor(mx, 8));
            float mnew  = fmaxf(mrow[r], mx);
            float alpha = __expf(mrow[r] - mnew);
            float p     = __expf(sv - mnew);
            float ps = p;
            ps += __shfl_xor(ps, 1);
            ps += __shfl_xor(ps, 2);
            ps += __shfl_xor(ps, 4);
            ps += __shfl_xor(ps, 8);
            lrow[r] = lrow[r] * alpha + ps;
            mrow[r] = mnew;
            o0[r] *= alpha;
            o1[r] *= alpha;
            lp[m * LP_S + lo] = p;   // C-layout -> LDS [row][col]
        }
#pragma unroll
        for (int t = 0; t < 4; ++t) {
            v2f pa = *(const v2f*)&lp[lo * LP_S + t * 4 + klo];
            int krow = kj * 16 + t * 4 + klo;
            v2f vb0, vb1;
            vb0.x = V[(size_t)krow * 32 + lo];
            vb0.y = V[(size_t)(krow + 1) * 32 + lo];
            vb1.x = V[(size_t)krow * 32 + 16 + lo];
            vb1.y = V[(size_t)(krow + 1) * 32 + 16 + lo];
            o0 = wmma_f32_16x16x4(pa, vb0, o0);
            o1 = wmma_f32_16x16x4(pa, vb1, o1);
        }
    }

    const int b = bh >> 6, h = bh & 63;
#pragma unroll
    for (int r = 0; r < 8; ++r) {
        int q = qi * 16 + r + hi * 8;
        float inv = 1.0f / lrow[r];
        size_t rowbase = ((size_t)(b * 1024 + q)) * 2048 + h * 32;
        y[rowbase + lo]      = o0[r] * inv;
        y[rowbase + 16 + lo] = o1[r] * inv;
    }
}

// ---------------------------------------------------------------------------
extern "C" void kernel_launch(void* const* d_in, const int* in_sizes, int n_in,
                              void* d_out, int out_size, void* d_ws, size_t ws_size,
                              hipStream_t stream) {
    const float* x      = (const float*)d_in[0];   // [2,1024,2048]
    const float* pos    = (const float*)d_in[1];   // [1024]
    const float* w_attn = (const float*)d_in[2];   // [2048,6144]
    const float* w_proj = (const float*)d_in[3];   // [2048,2048]
    float* out = (float*)d_out;                    // [2,1024,2048]

    const size_t HTD = (size_t)2048 * 2048;
    float* qbuf = (float*)d_ws;                    // [B,H,T,32]
    float* kbuf = qbuf + HTD;
    float* vbuf = kbuf + HTD;
    float* ybuf = vbuf + HTD;                      // [B*T, 2048]

    // 1) QKV projection, scatter into [B,H,T,hd] q/k/v buffers.
    dim3 g1(2048 / 64, 6144 / 128);
    attn_gemm_f32_wmma<<<g1, 256, 0, stream>>>(
        x, w_attn, nullptr, 2048, 6144, 2048, 1, qbuf, kbuf, vbuf);

    // 2) RoPE in place on q and k.
    const int pairs = 2 * 64 * 1024 * 16;
    rope_inplace<<<(pairs + 255) / 256, 256, 0, stream>>>(qbuf, kbuf, pos);

    // 3) Flash attention -> y.
    flash_attn_f32_wmma<<<2048, 128, 0, stream>>>(qbuf, kbuf, vbuf, ybuf);

    // 4) Output projection.
    dim3 g2(2048 / 64, 2048 / 128);
    attn_gemm_f32_wmma<<<g2, 256, 0, stream>>>(
        ybuf, w_proj, out, 2048, 2048, 2048, 0, nullptr, nullptr, nullptr);
}